// SAGE_35115652612101
// MI455X (gfx1250) — compile-verified
//
#include <hip/hip_runtime.h>
#include <math.h>

#define SAGE_IN_CH 128

typedef float v2f __attribute__((ext_vector_type(2)));
typedef float v8f __attribute__((ext_vector_type(8)));

// ---------------------------------------------------------------------------
// Zero an int/float buffer (0 bits). d_ws is poisoned, must init ourselves.
// ---------------------------------------------------------------------------
__global__ void sage_zero_b32(int* __restrict__ p, long n) {
    long i = (long)blockIdx.x * blockDim.x + threadIdx.x;
    if (i < n) p[i] = 0;
}

// ---------------------------------------------------------------------------
// CSR build step 1: in-degree count (int atomics, one-time cost)
// ---------------------------------------------------------------------------
__global__ void sage_degree(const int* __restrict__ dst, int* __restrict__ cnt, int E) {
    int e = blockIdx.x * blockDim.x + threadIdx.x;
    if (e < E) atomicAdd(&cnt[dst[e]], 1);
}

// ---------------------------------------------------------------------------
// CSR build step 2: exclusive prefix scan (single block, chunked Hillis-Steele
// in LDS with running carry). n = 50000 -> ~196 chunks, negligible runtime.
// ---------------------------------------------------------------------------
__global__ void sage_scan(const int* __restrict__ cnt, int* __restrict__ rowstart, int n) {
    __shared__ int buf[256];
    __shared__ int carry;
    const int tid = threadIdx.x;
    if (tid == 0) carry = 0;
    __syncthreads();
    for (int base = 0; base < n; base += 256) {
        int i = base + tid;
        int v = (i < n) ? cnt[i] : 0;
        buf[tid] = v;
        __syncthreads();
        for (int off = 1; off < 256; off <<= 1) {
            int t = (tid >= off) ? buf[tid - off] : 0;
            __syncthreads();
            buf[tid] += t;
            __syncthreads();
        }
        int c = carry;
        if (i < n) rowstart[i] = c + buf[tid] - v;   // exclusive
        __syncthreads();
        if (tid == 255) carry += buf[255];
        __syncthreads();
    }
    if (tid == 0) rowstart[n] = carry;               // == E
}

// ---------------------------------------------------------------------------
// CSR build step 3: bucket-fill neighbor lists
// ---------------------------------------------------------------------------
__global__ void sage_fill(const int* __restrict__ src, const int* __restrict__ dst,
                          const int* __restrict__ rowstart, int* __restrict__ cursor,
                          int* __restrict__ nbr, int E) {
    int e = blockIdx.x * blockDim.x + threadIdx.x;
    if (e >= E) return;
    int d = dst[e];
    int p = atomicAdd(&cursor[d], 1);
    nbr[rowstart[d] + p] = src[e];
}

// ---------------------------------------------------------------------------
// Gather-mean (replaces scatter-atomics entirely): one wave32 per dst node,
// each lane owns 4 consecutive channels (float4 -> global_load_b128,
// coalesced 512B per wave per neighbor). Loop bounds are wave-uniform.
// ---------------------------------------------------------------------------
__global__ void sage_gather_mean(const float* __restrict__ x,
                                 const int* __restrict__ rowstart,
                                 const int* __restrict__ nbr,
                                 float* __restrict__ mean, int N) {
    int node = blockIdx.x * (blockDim.x >> 5) + (threadIdx.x >> 5);
    if (node >= N) return;
    const int lane = threadIdx.x & 31;
    const int beg = rowstart[node];
    const int end = rowstart[node + 1];
    float a0 = 0.f, a1 = 0.f, a2 = 0.f, a3 = 0.f;
    for (int j = beg; j < end; ++j) {
        int s = nbr[j];
        const float4 v = *(const float4*)(x + (size_t)s * SAGE_IN_CH + lane * 4);
        a0 += v.x; a1 += v.y; a2 += v.z; a3 += v.w;
    }
    const float invd = 1.0f / fmaxf((float)(end - beg), 1.0f);
    float4 r; r.x = a0 * invd; r.y = a1 * invd; r.z = a2 * invd; r.w = a3 * invd;
    *(float4*)(mean + (size_t)node * SAGE_IN_CH + lane * 4) = r;
}

// ---------------------------------------------------------------------------
// Fused SAGE layer GEMM:  out = mean @ Wl + x @ Wr + b  [+ ELU]
// Block = 8 waves = 8 M-tiles sharing one N-tile. The 128x16 B-tiles of Wl/Wr
// are cooperatively staged in LDS (16KB) once per block; inner loop is
// ds_load (B) + global b64 (A) + V_WMMA_F32_16X16X4_F32 (fp32 in/out matches
// the fp32 reference). EXEC stays all-ones: tile guard is wave-uniform.
//
// A-fragment (16x4 f32): lane l -> row M=l&15; VGPR0 K=k0+2*(l>>4), VGPR1 K+1.
// B-fragment (4x16): lane l -> col N=l&15, same K striping (LDS banks:
// lanes 0-15 and 16-31 hit disjoint 16-word groups -> conflict-free).
// C/D (16x16 f32): VGPR v: lanes 0-15 -> M=v, lanes 16-31 -> M=v+8; N=l&15.
// ---------------------------------------------------------------------------
__global__ void sage_gemm_wmma(const float* __restrict__ mean,  // [N,128]
                               const float* __restrict__ xin,   // [N,128]
                               const float* __restrict__ Wl,    // [128,dout]
                               const float* __restrict__ Wr,    // [128,dout]
                               const float* __restrict__ bias,  // [dout]
                               float* __restrict__ out,         // [N,dout]
                               int n_nodes, int dout, int tilesM, int tilesN,
                               int apply_elu) {
    __shared__ float ldsL[SAGE_IN_CH * 16];
    __shared__ float ldsR[SAGE_IN_CH * 16];

    const int tileN   = blockIdx.x % tilesN;
    const int groupM  = blockIdx.x / tilesN;
    const int waveId  = threadIdx.x >> 5;
    const int lane    = threadIdx.x & 31;
    const int colBase = tileN * 16;

    // Cooperative stage: 128x16 tile of each weight matrix -> LDS
    #pragma unroll
    for (int i = 0; i < (SAGE_IN_CH * 16) / 256; ++i) {
        int idx = threadIdx.x + i * 256;
        int k = idx >> 4, n = idx & 15;
        ldsL[idx] = Wl[(size_t)k * dout + colBase + n];
        ldsR[idx] = Wr[(size_t)k * dout + colBase + n];
    }
    __syncthreads();

    const int tileM = groupM * 8 + waveId;
    if (tileM >= tilesM) return;                 // wave-uniform, after barrier

    const int mloc = lane & 15;
    const int hi   = lane >> 4;                  // 0 or 1
    const int koff = hi * 2;                     // K sub-offset within step of 4
    int rowA = tileM * 16 + mloc;
    if (rowA > n_nodes - 1) rowA = n_nodes - 1;  // branchless clamp
    const int col = colBase + mloc;

    const float* aM = mean + (size_t)rowA * SAGE_IN_CH + koff;
    const float* aX = xin + (size_t)rowA * SAGE_IN_CH + koff;

    v8f acc = {};
    #pragma unroll 4
    for (int k0 = 0; k0 < SAGE_IN_CH; k0 += 4) {
        v2f a, b;
        // mean-neighbors @ W_l
        a = *(const v2f*)(aM + k0);              // 8B aligned (koff even)
        b[0] = ldsL[(k0 + koff) * 16 + mloc];
        b[1] = ldsL[(k0 + koff + 1) * 16 + mloc];
        acc = __builtin_amdgcn_wmma_f32_16x16x4_f32(false, a, false, b,
                                                    (short)0, acc, false, false);
        // self @ W_r
        a = *(const v2f*)(aX + k0);
        b[0] = ldsR[(k0 + koff) * 16 + mloc];
        b[1] = ldsR[(k0 + koff + 1) * 16 + mloc];
        acc = __builtin_amdgcn_wmma_f32_16x16x4_f32(false, a, false, b,
                                                    (short)0, acc, false, false);
    }

    const float bv = bias[col];
    const int rowBase = tileM * 16 + (hi << 3);
    #pragma unroll
    for (int v = 0; v < 8; ++v) {
        int row = rowBase + v;
        float val = acc[v] + bv;
        if (apply_elu) val = (val > 0.0f) ? val : expm1f(val);
        if (row < n_nodes) out[(size_t)row * dout + col] = val;
    }
}

// ---------------------------------------------------------------------------
// Host-side orchestration
// ---------------------------------------------------------------------------
static void run_layer(const float* xin, const int* rowstart, const int* nbr,
                      const float* Wl, const float* Wr, const float* b,
                      float* mean, float* out,
                      int N, int dout, int apply_elu, hipStream_t stream) {
    int gblocks = (N + 7) / 8;                   // one wave per node, 8 waves/block
    sage_gather_mean<<<gblocks, 256, 0, stream>>>(xin, rowstart, nbr, mean, N);

    int tilesM = (N + 15) / 16;
    int tilesN = dout / 16;
    int blocks = ((tilesM + 7) / 8) * tilesN;
    sage_gemm_wmma<<<blocks, 256, 0, stream>>>(mean, xin, Wl, Wr, b, out,
                                               N, dout, tilesM, tilesN, apply_elu);
}

extern "C" void kernel_launch(void* const* d_in, const int* in_sizes, int n_in,
                              void* d_out, int out_size, void* d_ws, size_t ws_size,
                              hipStream_t stream) {
    const float* x    = (const float*)d_in[0];
    const int*   edge = (const int*)d_in[1];     // [2, E]
    const float* Wl0  = (const float*)d_in[2];
    const float* Wr0  = (const float*)d_in[3];
    const float* b0   = (const float*)d_in[4];
    const float* Wl1  = (const float*)d_in[5];
    const float* Wr1  = (const float*)d_in[6];
    const float* b1   = (const float*)d_in[7];
    const float* Wl2  = (const float*)d_in[8];
    const float* Wr2  = (const float*)d_in[9];
    const float* b2   = (const float*)d_in[10];

    const int N = in_sizes[0] / SAGE_IN_CH;      // 50000
    const int E = in_sizes[1] / 2;               // 800000
    const int* src = edge;
    const int* dst = edge + E;

    // Workspace layout (4-byte units): cnt/cursor | rowstart | nbr | mean | h1 | h2
    int* ws       = (int*)d_ws;
    int* cursor   = ws;                          // N   (reused: cnt, then fill cursor)
    int* rowstart = cursor + N;                  // N+1
    int* nbr      = rowstart + N + 1;            // E
    size_t fbase  = (size_t)(2 * N + 1 + E);
    fbase = (fbase + 3) & ~(size_t)3;            // 16B-align the float region
    float* mean   = (float*)ws + fbase;          // N*128
    float* h1     = mean + (size_t)N * SAGE_IN_CH;
    float* h2     = h1 + (size_t)N * SAGE_IN_CH;

    // --- CSR build (once; graph identical across all 3 layers) ---
    sage_zero_b32<<<(N + 255) / 256, 256, 0, stream>>>(cursor, N);
    sage_degree<<<(E + 255) / 256, 256, 0, stream>>>(dst, cursor, E);
    sage_scan<<<1, 256, 0, stream>>>(cursor, rowstart, N);
    sage_zero_b32<<<(N + 255) / 256, 256, 0, stream>>>(cursor, N);
    sage_fill<<<(E + 255) / 256, 256, 0, stream>>>(src, dst, rowstart, cursor, nbr, E);

    // --- Layers ---
    run_layer(x,  rowstart, nbr, Wl0, Wr0, b0, mean, h1, N, 128, 1, stream);
    run_layer(h1, rowstart, nbr, Wl1, Wr1, b1, mean, h2, N, 128, 1, stream);
    run_layer(h2, rowstart, nbr, Wl2, Wr2, b2, mean, (float*)d_out, N, 64, 0, stream);
}